// Net_76270029242478
// MI455X (gfx1250) — compile-verified
//
#include <hip/hip_runtime.h>

typedef float v2f __attribute__((ext_vector_type(2)));
typedef float v8f __attribute__((ext_vector_type(8)));

#define B_   32
#define N_   100
#define E_   256
#define H_   1024
#define MID_ 256

__device__ __forceinline__ v8f wmma_f32_16x16x4(v2f a, v2f b, v8f c) {
  // D = A(16x4 f32) x B(4x16 f32) + C(16x16 f32)
  return __builtin_amdgcn_wmma_f32_16x16x4_f32(false, a, false, b, (short)0, c,
                                               false, false);
}

// ---------------------------------------------------------------------------
// Kernel 1: s[b,m] = (q @ W_q + b_q)[b,m] * W_out[m]
// GEMM M=32 (b), N=256 (m), K=1024 (h). 2x16 tiles, one wave per 16x16 tile.
// ---------------------------------------------------------------------------
__global__ __launch_bounds__(32) void k_qproj(const float* __restrict__ q,
                                              const float* __restrict__ W_q,
                                              const float* __restrict__ b_q,
                                              const float* __restrict__ W_out,
                                              float* __restrict__ s) {
  const int lane = threadIdx.x;
  const int half = lane >> 4, lr = lane & 15;
  const int t = blockIdx.x;
  const int mtile = t & 1;    // b-row tile (2 tiles of 16)
  const int ntile = t >> 1;   // m-col tile (16 tiles of 16)
  const int brow = mtile * 16 + lr;
  const int mcol = ntile * 16 + lr;
  const float* qrow = q + (size_t)brow * H_;
  v8f acc = {};
  for (int kk = 0; kk < H_ / 4; ++kk) {
    const int k = kk * 4 + 2 * half;            // lane's K slot per ISA layout
    v2f a = *(const v2f*)(qrow + k);            // A[M=lr][k,k+1]
    v2f b;                                      // B[k,k+1][N=lr] (column of W_q)
    b.x = W_q[(size_t)k * MID_ + mcol];
    b.y = W_q[(size_t)(k + 1) * MID_ + mcol];
    acc = wmma_f32_16x16x4(a, b, acc);
  }
  const float bq = b_q[mcol], wo = W_out[mcol];
  const int rb = mtile * 16 + 8 * half;         // D row = r + 8*half
#pragma unroll
  for (int r = 0; r < 8; ++r)
    s[(size_t)(rb + r) * MID_ + mcol] = (acc[r] + bq) * wo;
}

// ---------------------------------------------------------------------------
// Kernel 2: v[b,e] = sum_m W_r[e,m] * s[b,m]
// GEMM M=256 (e), N=32 (b), K=256 (m). 16x2 tiles, one wave per tile.
// ---------------------------------------------------------------------------
__global__ __launch_bounds__(32) void k_vmat(const float* __restrict__ W_r,
                                             const float* __restrict__ s,
                                             float* __restrict__ v) {
  const int lane = threadIdx.x;
  const int half = lane >> 4, lr = lane & 15;
  const int t = blockIdx.x;
  const int et = t & 15;   // 16 e-tiles
  const int bt = t >> 4;   // 2 b-tiles
  const int erow = et * 16 + lr;
  const int bcol = bt * 16 + lr;
  const float* arow = W_r + (size_t)erow * MID_;   // A[e][m], contiguous in m
  const float* brow = s + (size_t)bcol * MID_;     // B[m][b] = s[b][m]
  v8f acc = {};
  for (int kk = 0; kk < MID_ / 4; ++kk) {
    const int k = kk * 4 + 2 * half;
    v2f a = *(const v2f*)(arow + k);
    v2f b = *(const v2f*)(brow + k);
    acc = wmma_f32_16x16x4(a, b, acc);
  }
  const int eb = et * 16 + 8 * half;
#pragma unroll
  for (int r = 0; r < 8; ++r)
    v[(size_t)bcol * E_ + (eb + r)] = acc[r];      // store as v[b][e]
}

// ---------------------------------------------------------------------------
// Kernel 3: c[b] = dot(b_r, s[b]) + b_out
// ---------------------------------------------------------------------------
__global__ __launch_bounds__(32) void k_cvec(const float* __restrict__ b_r,
                                             const float* __restrict__ s,
                                             const float* __restrict__ b_out,
                                             float* __restrict__ c) {
  const int b = threadIdx.x;
  float acc = 0.f;
  for (int m = 0; m < MID_; ++m) acc += b_r[m] * s[b * MID_ + m];
  c[b] = acc + b_out[0];
}

// ---------------------------------------------------------------------------
// Kernel 4 (the 327.7 MB streaming pass, HBM-bound):
// relation[b,i,j] = dot(trans_mat[b,i,j,:], v[b,:]) + c[b]
// One wave per 16 rows; K=4 per WMMA; B operand broadcasts v across all 16
// columns so every D column holds the same dot -> lanes 0/16 own rows 0-7/8-15
// contiguously. 625 tiles per batch; 5 waves/block -> 125 blocks/batch.
// ---------------------------------------------------------------------------
__global__ __launch_bounds__(160) void k_relation(const float* __restrict__ tm,
                                                  const float* __restrict__ v,
                                                  const float* __restrict__ c,
                                                  float* __restrict__ rel) {
  __shared__ float vsh[E_];
  const int tid = threadIdx.x;
  const int b = blockIdx.x / 125;
  const int blkInBatch = blockIdx.x % 125;
  for (int i = tid; i < E_; i += 160) vsh[i] = v[b * E_ + i];
  __syncthreads();

  const int wave = tid >> 5, lane = tid & 31;
  const int half = lane >> 4, lr = lane & 15;
  const int tile = blkInBatch * 5 + wave;                 // 0..624
  const size_t rowBase = (size_t)b * (N_ * N_) + (size_t)tile * 16;
  const float* arow = tm + (rowBase + lr) * E_;           // this lane's row

  v8f acc0 = {}, acc1 = {};
  for (int kk = 0; kk < E_ / 4; kk += 2) {                // 2 accumulators
    const int k0 = kk * 4 + 2 * half;
    const int k1 = k0 + 4;
    v2f a0 = *(const v2f*)(arow + k0);
    v2f b0 = *(const v2f*)(vsh + k0);                     // broadcast to all N
    v2f a1 = *(const v2f*)(arow + k1);
    v2f b1 = *(const v2f*)(vsh + k1);
    acc0 = wmma_f32_16x16x4(a0, b0, acc0);
    acc1 = wmma_f32_16x16x4(a1, b1, acc1);
  }
  if (lr == 0) {                                          // lanes 0 and 16
    const float cb = c[b];
    float* out = rel + rowBase + 8 * half;                // rows 0-7 / 8-15
#pragma unroll
    for (int r = 0; r < 8; ++r) out[r] = acc0[r] + acc1[r] + cb;
  }
}

// ---------------------------------------------------------------------------
// Kernel 5: per-(b,j) masked column max & sum(exp) over i  (softmax axis=1)
// ---------------------------------------------------------------------------
__global__ void k_colstats(const float* __restrict__ rel,
                           const int* __restrict__ mask,
                           float* __restrict__ colmax,
                           float* __restrict__ colsum) {
  const int idx = blockIdx.x * blockDim.x + threadIdx.x;
  if (idx >= B_ * N_) return;
  const int b = idx / N_, j = idx % N_;
  const float* rbase = rel + (size_t)b * N_ * N_ + j;
  const int* mbase = mask + (size_t)b * N_ * N_ + j;
  float mx = -3.4e38f;
  for (int i = 0; i < N_; ++i)
    if (mbase[i * N_] != 0) mx = fmaxf(mx, rbase[i * N_]);
  float sum = 0.f;
  for (int i = 0; i < N_; ++i)
    if (mbase[i * N_] != 0) sum += __expf(rbase[i * N_] - mx);
  colmax[idx] = mx;
  colsum[idx] = sum;
}

// ---------------------------------------------------------------------------
// Kernel 6: new_logits[b,i] = sum_j softmax_col(relation)[b,i,j] * z[b,j]
// (row i of relation is contiguous in j -> coalesced)
// ---------------------------------------------------------------------------
__global__ void k_aggregate(const float* __restrict__ rel,
                            const int* __restrict__ mask,
                            const float* __restrict__ colmax,
                            const float* __restrict__ colsum,
                            const float* __restrict__ z,
                            float* __restrict__ out) {
  const int idx = blockIdx.x * blockDim.x + threadIdx.x;
  if (idx >= B_ * N_) return;
  const int b = idx / N_, i = idx % N_;
  const float* rbase = rel + ((size_t)b * N_ + i) * N_;
  const int* mbase = mask + ((size_t)b * N_ + i) * N_;
  float acc = 0.f;
  for (int j = 0; j < N_; ++j) {
    if (mbase[j] != 0) {
      const float den = colsum[b * N_ + j];
      const float w = __expf(rbase[j] - colmax[b * N_ + j]) / den;
      acc += w * z[b * N_ + j];
    }
  }
  out[idx] = acc;
}

// ---------------------------------------------------------------------------
extern "C" void kernel_launch(void* const* d_in, const int* in_sizes, int n_in,
                              void* d_out, int out_size, void* d_ws, size_t ws_size,
                              hipStream_t stream) {
  const float* q      = (const float*)d_in[0];
  const float* tm     = (const float*)d_in[1];
  const int*   r_mask = (const int*)d_in[2];
  const float* z      = (const float*)d_in[3];
  const float* W_r    = (const float*)d_in[4];
  const float* b_r    = (const float*)d_in[5];
  const float* W_q    = (const float*)d_in[6];
  const float* b_q    = (const float*)d_in[7];
  const float* W_out  = (const float*)d_in[8];
  const float* b_out  = (const float*)d_in[9];
  float* out = (float*)d_out;

  float* ws     = (float*)d_ws;
  float* s      = ws;            // 8192 floats
  float* v      = ws + 8192;     // 8192
  float* c      = ws + 16384;    // 32 (padded)
  float* colmax = ws + 16512;    // 3200
  float* colsum = ws + 19712;    // 3200 (padded to 23040)
  float* rel    = ws + 23040;    // 320000  -> total ~1.37 MB

  k_qproj<<<dim3(32), dim3(32), 0, stream>>>(q, W_q, b_q, W_out, s);
  k_vmat<<<dim3(32), dim3(32), 0, stream>>>(W_r, s, v);
  k_cvec<<<dim3(1), dim3(32), 0, stream>>>(b_r, s, b_out, c);
  k_relation<<<dim3(4000), dim3(160), 0, stream>>>(tm, v, c, rel);
  const int nt = B_ * N_;
  k_colstats<<<dim3((nt + 255) / 256), dim3(256), 0, stream>>>(rel, r_mask, colmax, colsum);
  k_aggregate<<<dim3((nt + 255) / 256), dim3(256), 0, stream>>>(rel, r_mask, colmax, colsum, z, out);
}